// BiologicalModule_38165079392670
// MI455X (gfx1250) — compile-verified
//
#include <hip/hip_runtime.h>
#include <math.h>

#define INPUT_DIM 30000
#define UNITS     2048
#define NNZ       500000
#define BATCH     32
#define TILES     (UNITS / 16)   // 128 column tiles of width 16
#define SPLIT     8              // sub-blocks per tile for parallelism

typedef __attribute__((ext_vector_type(2))) float v2f;
typedef __attribute__((ext_vector_type(8))) float v8f;

// ---- zero histogram + accumulator ------------------------------------------
__global__ void k_init(unsigned* cnt, float* acc) {
    int t = blockIdx.x * blockDim.x + threadIdx.x;
    if (t < TILES) cnt[t] = 0u;
    int a = t - TILES;
    if (a >= 0 && a < BATCH * UNITS) acc[a] = 0.0f;
}

// ---- transpose x[32,30000] -> xT[30000,32] (L2-resident gather target) -----
__global__ void k_transpose(const float* __restrict__ x, float* __restrict__ xT) {
    int t = blockIdx.x * blockDim.x + threadIdx.x;
    if (t >= BATCH * INPUT_DIM) return;
    int b = t / INPUT_DIM;
    int r = t - b * INPUT_DIM;
    xT[r * BATCH + b] = x[t];   // coalesced read of x
}

// ---- histogram nonzeros per 16-wide column tile ----------------------------
__global__ void k_count(const int* __restrict__ ind, unsigned* cnt) {
    int i = blockIdx.x * blockDim.x + threadIdx.x;
    if (i >= NNZ) return;
    int col = ind[2 * i + 1];
    atomicAdd(&cnt[col >> 4], 1u);
}

// ---- tiny exclusive scan with bins padded to multiples of 8 entries --------
__global__ void k_scan(const unsigned* __restrict__ cnt, unsigned* offs, unsigned* cur) {
    if (blockIdx.x == 0 && threadIdx.x == 0) {
        unsigned run = 0;
        for (int i = 0; i < TILES; ++i) {
            offs[i] = run; cur[i] = run;
            run += (cnt[i] + 7u) & ~7u;   // 8-entry aligned bins (2 chunks)
        }
        offs[TILES] = run;
    }
}

// ---- scatter nonzeros into tile bins: pack (row[15b] | colmod[4b], val) ----
__global__ void k_scatter(const int* __restrict__ ind, const float* __restrict__ kv,
                          unsigned* cur, uint2* __restrict__ packed) {
    int i = blockIdx.x * blockDim.x + threadIdx.x;
    if (i >= NNZ) return;
    int row = ind[2 * i];
    int col = ind[2 * i + 1];
    unsigned p = atomicAdd(&cur[col >> 4], 1u);
    uint2 e;
    e.x = (unsigned)row | ((unsigned)(col & 15) << 15);  // row < 30000 < 2^15
    e.y = __float_as_uint(kv[i]);
    packed[p] = e;
}

// ---- zero the <=7 pad slots at the end of each bin (val=0 -> WMMA no-op) ---
__global__ void k_pad(const unsigned* __restrict__ cnt, const unsigned* __restrict__ offs,
                      uint2* __restrict__ packed) {
    int t = blockIdx.x * blockDim.x + threadIdx.x;   // TILES*8 threads
    if (t >= TILES * 8) return;
    int tile = t >> 3;
    unsigned slot = offs[tile] + cnt[tile] + (unsigned)(t & 7);
    if (slot < offs[tile + 1]) packed[slot] = make_uint2(0u, 0u);
}

// ---- gather-GEMM: 4 nonzeros per V_WMMA_F32_16X16X4_F32 --------------------
// Block = 64 threads = 2 waves; wave w owns batch rows [16w, 16w+16).
// A[16,4] = xT[row_k][batch], B[4,16] = one-hot(col_k mod 16)*val_k.
// Bins padded to 8 entries; per-sub chunk count forced even -> unroll x2 with
// two independent accumulators, no bounds checks, no branches, EXEC all-ones.
__global__ void k_spmm_wmma(const float* __restrict__ xT, const uint4* __restrict__ pack4,
                            const unsigned* __restrict__ offs, float* __restrict__ acc) {
    const int tile  = blockIdx.x / SPLIT;
    const int sub   = blockIdx.x % SPLIT;
    const int lane  = threadIdx.x & 31;
    const int wave  = threadIdx.x >> 5;     // batch half
    const int batchBase = wave * 16;
    const unsigned m     = (unsigned)(lane & 15); // A row (M) / B,C col (N)
    const int      khalf = lane >> 4;       // K-pair: lanes 0-15 -> K{0,1}, 16-31 -> K{2,3}
    const unsigned beg = offs[tile];
    const unsigned end = offs[tile + 1];
    const unsigned nchunk = (end - beg) >> 2;              // exact, even (8-padded)
    unsigned per = (nchunk + (unsigned)SPLIT - 1u) / (unsigned)SPLIT;
    per = (per + 1u) & ~1u;                                // even split -> pairs only
    unsigned c0 = (unsigned)sub * per;
    unsigned c1 = c0 + per;
    if (c0 > nchunk) c0 = nchunk;
    if (c1 > nchunk) c1 = nchunk;

    v8f Ca = {}, Cb = {};
    for (unsigned c = c0; c < c1; c += 2u) {  // uniform bounds -> EXEC all ones
        unsigned q = (beg >> 1) + (c << 1) + (unsigned)khalf; // uint4 index
        uint4 e0 = pack4[q];                 // chunk c,   entries 2*khalf..+1
        uint4 e1 = pack4[q + 2u];            // chunk c+1, entries 2*khalf..+1
        // chunk c
        {
            unsigned r0 = e0.x & 0x7FFFu, r1 = e0.z & 0x7FFFu;
            unsigned k0 = 0u - (unsigned)(((e0.x >> 15) & 0xFu) == m);
            unsigned k1 = 0u - (unsigned)(((e0.z >> 15) & 0xFu) == m);
            v2f A, B;
            A.x = xT[r0 * BATCH + (unsigned)batchBase + m];
            A.y = xT[r1 * BATCH + (unsigned)batchBase + m];
            B.x = __uint_as_float(e0.y & k0);   // branchless one-hot
            B.y = __uint_as_float(e0.w & k1);
            Ca = __builtin_amdgcn_wmma_f32_16x16x4_f32(false, A, false, B,
                                                       (short)0, Ca, false, false);
        }
        // chunk c+1 (independent accumulator -> WMMAs can overlap)
        {
            unsigned r0 = e1.x & 0x7FFFu, r1 = e1.z & 0x7FFFu;
            unsigned k0 = 0u - (unsigned)(((e1.x >> 15) & 0xFu) == m);
            unsigned k1 = 0u - (unsigned)(((e1.z >> 15) & 0xFu) == m);
            v2f A, B;
            A.x = xT[r0 * BATCH + (unsigned)batchBase + m];
            A.y = xT[r1 * BATCH + (unsigned)batchBase + m];
            B.x = __uint_as_float(e1.y & k0);
            B.y = __uint_as_float(e1.w & k1);
            Cb = __builtin_amdgcn_wmma_f32_16x16x4_f32(false, A, false, B,
                                                       (short)0, Cb, false, false);
        }
    }

    // Merge partial C tile: VGPR r holds M = r + 8*khalf, N = lane&15
    const int rowOff = khalf * 8;
    #pragma unroll
    for (int r = 0; r < 8; ++r) {
        int mm = r + rowOff;
        atomicAdd(&acc[(batchBase + mm) * UNITS + tile * 16 + (int)m], Ca[r] + Cb[r]);
    }
}

// ---- bias + tanh epilogue ---------------------------------------------------
__global__ void k_final(const float* __restrict__ acc, const float* __restrict__ bias,
                        float* __restrict__ out) {
    int t = blockIdx.x * blockDim.x + threadIdx.x;
    if (t >= BATCH * UNITS) return;
    int c = t & (UNITS - 1);
    out[t] = tanhf(acc[t] + bias[c]);
}

// ---- workspace layout (bytes) ----------------------------------------------
//   xT   : [0, 3,840,000)           30000*32 f32
//   cnt  : [3,840,000, +512)        128 u32
//   offs : [3,840,512, +516)        129 u32
//   cur  : [3,841,152, +512)        128 u32
//   acc  : [3,841,664, +262,144)    32*2048 f32
//   pack : [4,103,808, +4,008,192)  (500000 + 1024) x uint2, 16B aligned
// total ~8.11 MB

extern "C" void kernel_launch(void* const* d_in, const int* in_sizes, int n_in,
                              void* d_out, int out_size, void* d_ws, size_t ws_size,
                              hipStream_t stream) {
    const float* x    = (const float*)d_in[0];
    const float* kv   = (const float*)d_in[1];
    const float* bias = (const float*)d_in[2];
    const int*   ind  = (const int*)d_in[3];
    float* out = (float*)d_out;

    char* ws = (char*)d_ws;
    float*    xT   = (float*)(ws + 0);
    unsigned* cnt  = (unsigned*)(ws + 3840000);
    unsigned* offs = (unsigned*)(ws + 3840512);
    unsigned* cur  = (unsigned*)(ws + 3841152);
    float*    acc  = (float*)(ws + 3841664);
    uint2*    pack = (uint2*)(ws + 4103808);

    k_init<<<(BATCH * UNITS + TILES + 255) / 256, 256, 0, stream>>>(cnt, acc);
    k_transpose<<<(BATCH * INPUT_DIM + 255) / 256, 256, 0, stream>>>(x, xT);
    k_count<<<(NNZ + 255) / 256, 256, 0, stream>>>(ind, cnt);
    k_scan<<<1, 32, 0, stream>>>(cnt, offs, cur);
    k_scatter<<<(NNZ + 255) / 256, 256, 0, stream>>>(ind, kv, cur, pack);
    k_pad<<<(TILES * 8 + 255) / 256, 256, 0, stream>>>(cnt, offs, pack);
    k_spmm_wmma<<<TILES * SPLIT, 64, 0, stream>>>(xT, (const uint4*)pack, offs, acc);
    k_final<<<(BATCH * UNITS + 255) / 256, 256, 0, stream>>>(acc, bias, out);
}